// CurvatureGatedMessagePropagationLayer_78030965834356
// MI455X (gfx1250) — compile-verified
//
#include <hip/hip_runtime.h>
#include <hip/hip_bf16.h>

#define DIM 64
#define TAUF 0.5f
#define EPSF 1e-12f
#define SURGF 1e-6f

typedef float v2f __attribute__((ext_vector_type(2)));
typedef float v8f __attribute__((ext_vector_type(8)));

// Non-returning hardware fp32 atomic add at device scope: resolves in L2,
// tracked on STOREcnt (no wave stall waiting for a return value).
__device__ __forceinline__ void atomAddF(float* p, float v) {
    asm volatile("global_atomic_add_f32 %0, %1, off scope:SCOPE_DEV"
                 :: "v"(p), "v"(v) : "memory");
}

// ---------------- zero accumulators ----------------
__global__ void k_zero(float* __restrict__ ws, long n) {
    long i = (long)blockIdx.x * blockDim.x + threadIdx.x;
    if (i < n) ws[i] = 0.0f;
}

// ---------------- weighted out-degree: deg[row] += w ----------------
__global__ void k_deg(const long long* __restrict__ row, const float* __restrict__ w,
                      float* __restrict__ deg, int E) {
    int e = blockIdx.x * blockDim.x + threadIdx.x;
    if (e < E) atomAddF(deg + (int)row[e], w[e]);
}

// ---------------- kappa per edge + global max|kappa| ----------------
__global__ void k_kappa(const long long* __restrict__ row, const long long* __restrict__ col,
                        const float* __restrict__ deg, float* __restrict__ kappa,
                        unsigned* __restrict__ maxabs, int E) {
    __shared__ float red[256];
    int e = blockIdx.x * blockDim.x + threadIdx.x;
    float ak = 0.0f;
    if (e < E) {
        float du = fmaxf(deg[(int)row[e]], EPSF);
        float dv = fmaxf(deg[(int)col[e]], EPSF);
        float k = 2.0f / du + 2.0f / dv - 2.0f;
        kappa[e] = k;
        ak = fabsf(k);
    }
    red[threadIdx.x] = ak;
    __syncthreads();
    for (int s = 128; s > 0; s >>= 1) {
        if (threadIdx.x < (unsigned)s)
            red[threadIdx.x] = fmaxf(red[threadIdx.x], red[threadIdx.x + s]);
        __syncthreads();
    }
    if (threadIdx.x == 0) atomicMax(maxabs, __float_as_uint(red[0]));  // |k| >= 0: uint order == float order
}

// ---------------- Ricci-flow half step + surgery + node sums ----------------
__global__ void k_flow(const long long* __restrict__ row, const float* __restrict__ w,
                       const float* __restrict__ kappa, const unsigned* __restrict__ maxabs,
                       float* __restrict__ w_half, float* __restrict__ row_sum,
                       float* __restrict__ cnt, float* __restrict__ ksum, int E) {
    int e = blockIdx.x * blockDim.x + threadIdx.x;
    if (e >= E) return;
    float dt = 0.5f / (__uint_as_float(*maxabs) + 1e-6f);
    float k = kappa[e];
    float wh = fmaxf(w[e] * (1.0f - 0.5f * dt * k), 0.0f);
    bool m = wh > SURGF;
    wh = m ? wh : 0.0f;
    w_half[e] = wh;
    if (m) {
        int r = (int)row[e];
        atomAddF(row_sum + r, wh);
        atomAddF(cnt + r, 1.0f);
        atomAddF(ksum + r, k);
    }
}

// ---------------- curvature gate rho = sigmoid(mean kappa) ----------------
__global__ void k_rho(const float* __restrict__ cnt, const float* __restrict__ ksum,
                      float* __restrict__ rho, int N) {
    int n = blockIdx.x * blockDim.x + threadIdx.x;
    if (n < N) {
        float mk = ksum[n] / fmaxf(cnt[n], 1.0f);
        rho[n] = 1.0f / (1.0f + expf(-mk));
    }
}

// ---------------- dual 64x64 GEMM via V_WMMA_F32_16X16X4_F32 ----------------
// y = x @ W.T  =>  D[m][n] = sum_k A[m][k] * B[k][n],  B[k][n] = W[n][k]
// out_base = rho * h_self + TAU * x   ;   hN = h_neigh (kept for scatter)
__global__ __launch_bounds__(128) void k_gemm(const float* __restrict__ x,
                                              const float* __restrict__ Wself,
                                              const float* __restrict__ Wneigh,
                                              const float* __restrict__ rho,
                                              float* __restrict__ outb,
                                              float* __restrict__ hN, int N) {
    __shared__ float lWs[64][65];   // pitch 65 breaks the stride-64 bank conflict
    __shared__ float lWn[64][65];
    int t = threadIdx.x;
    for (int i = t; i < 64 * 64; i += 128) {
        lWs[i >> 6][i & 63] = Wself[i];
        lWn[i >> 6][i & 63] = Wneigh[i];
    }
    __syncthreads();

    int wave = t >> 5;
    int lane = t & 31;
    int hi   = lane >> 4;      // 0: K+0/1 ; 1: K+2/3  (32-bit A layout)
    int l15  = lane & 15;
    int m0   = blockIdx.x * 64 + wave * 16;

    // A fragments for all 16 K-steps of this wave's 16-row tile
    int arow = m0 + l15;
    arow = arow < N ? arow : N - 1;          // clamp tail (stores are guarded)
    const float* xr = x + (long)arow * DIM + hi * 2;
    v2f a[16];
#pragma unroll
    for (int kk = 0; kk < 16; ++kk)
        a[kk] = *(const v2f*)(xr + kk * 4);  // 8B aligned (even offsets)

    for (int nt = 0; nt < 4; ++nt) {
        v8f cs = {};
        v8f cn = {};
        int nrow = nt * 16 + l15;            // B: lanes carry the N dim
#pragma unroll
        for (int kk = 0; kk < 16; ++kk) {
            int kb = kk * 4 + hi * 2;
            v2f bs = { lWs[nrow][kb], lWs[nrow][kb + 1] };
            v2f bn = { lWn[nrow][kb], lWn[nrow][kb + 1] };
            cs = __builtin_amdgcn_wmma_f32_16x16x4_f32(false, a[kk], false, bs,
                                                       (short)0, cs, false, false);
            cn = __builtin_amdgcn_wmma_f32_16x16x4_f32(false, a[kk], false, bn,
                                                       (short)0, cn, false, false);
        }
        // C/D layout: VGPR r, lanes 0-15 -> M=r ; lanes 16-31 -> M=8+r ; N=lane&15
#pragma unroll
        for (int r = 0; r < 8; ++r) {
            int m = m0 + r + hi * 8;
            if (m < N) {
                int ncol = nt * 16 + l15;
                long idx = (long)m * DIM + ncol;
                outb[idx] = rho[m] * cs[r] + TAUF * x[idx];
                hN[idx]   = cn[r];
            }
        }
    }
}

// ---------------- row-normalised weights (also final output) ----------------
__global__ void k_wnorm(const long long* __restrict__ row, const float* __restrict__ w_half,
                        const float* __restrict__ row_sum, float* __restrict__ wn, int E) {
    int e = blockIdx.x * blockDim.x + threadIdx.x;
    if (e < E) wn[e] = w_half[e] / fmaxf(row_sum[(int)row[e]], EPSF);
}

// ---------------- gather h_neigh[col], scale, scatter-add to out[row] ----------------
// 16 threads per edge, each owns 4 contiguous dims: one 128-bit load + 4 f32 atomics
__global__ void k_scatter(const long long* __restrict__ row, const long long* __restrict__ col,
                          const float* __restrict__ wn, const float* __restrict__ hN,
                          float* __restrict__ out, int E) {
    long idx = (long)blockIdx.x * blockDim.x + threadIdx.x;
    long tot = (long)E * 16;
    if (idx >= tot) return;
    int e   = (int)(idx >> 4);
    int sub = (int)(idx & 15);
    float w = wn[e];
    if (w == 0.0f) return;                   // surgically removed edge: contributes nothing
    int c = (int)col[e];
    int r = (int)row[e];
    float4 h = *(const float4*)(hN + (long)c * DIM + sub * 4);
    float* o = out + (long)r * DIM + sub * 4;
    atomAddF(o + 0, w * h.x);
    atomAddF(o + 1, w * h.y);
    atomAddF(o + 2, w * h.z);
    atomAddF(o + 3, w * h.w);
}

extern "C" void kernel_launch(void* const* d_in, const int* in_sizes, int n_in,
                              void* d_out, int out_size, void* d_ws, size_t ws_size,
                              hipStream_t stream) {
    const float*     x  = (const float*)d_in[0];
    const long long* ei = (const long long*)d_in[1];   // int64 [2, E]
    const float*     ew = (const float*)d_in[2];
    const float*     Ws = (const float*)d_in[3];
    const float*     Wn = (const float*)d_in[4];

    int N = in_sizes[0] / DIM;
    int E = in_sizes[2];
    const long long* row = ei;
    const long long* col = ei + E;

    // workspace layout (floats)
    float*    ws       = (float*)d_ws;
    float*    deg      = ws;
    float*    row_sum  = ws + (size_t)N;
    float*    cnt      = ws + 2 * (size_t)N;
    float*    ksum     = ws + 3 * (size_t)N;
    unsigned* maxabs   = (unsigned*)(ws + 4 * (size_t)N);   // zeroed with the block above
    float*    rho      = ws + 4 * (size_t)N + 8;
    float*    kappa    = rho + N;
    float*    w_half   = kappa + E;
    float*    hN       = w_half + E;        // N*DIM; 16B aligned only if 5N+2E is a multiple of 4

    // output layout: out [N*64 f32] | edge_index [2E i64] | w_norm [E f32]
    float*     outp   = (float*)d_out;
    long long* ei_out = (long long*)((char*)d_out + (size_t)N * DIM * sizeof(float));
    float*     wn_out = (float*)((char*)ei_out + (size_t)2 * E * sizeof(long long));

    long nz = 4 * (long)N + 1;
    k_zero  <<<(unsigned)((nz + 255) / 256), 256, 0, stream>>>(ws, nz);
    k_deg   <<<(E + 255) / 256, 256, 0, stream>>>(row, ew, deg, E);
    k_kappa <<<(E + 255) / 256, 256, 0, stream>>>(row, col, deg, kappa, maxabs, E);
    k_flow  <<<(E + 255) / 256, 256, 0, stream>>>(row, ew, kappa, maxabs, w_half,
                                                  row_sum, cnt, ksum, E);
    k_rho   <<<(N + 255) / 256, 256, 0, stream>>>(cnt, ksum, rho, N);
    k_gemm  <<<(N + 63) / 64, 128, 0, stream>>>(x, Ws, Wn, rho, outp, hN, N);
    k_wnorm <<<(E + 255) / 256, 256, 0, stream>>>(row, w_half, row_sum, wn_out, E);
    hipMemcpyAsync(ei_out, ei, (size_t)2 * E * sizeof(long long),
                   hipMemcpyDeviceToDevice, stream);
    long tot = (long)E * 16;
    k_scatter<<<(unsigned)((tot + 255) / 256), 256, 0, stream>>>(row, col, wn_out, hN, outp, E);
}